// Expert_WNO2d_21328807592355
// MI455X (gfx1250) — compile-verified
//
#include <hip/hip_runtime.h>
#include <hip/hip_bf16.h>

typedef __attribute__((ext_vector_type(2))) float v2f;
typedef __attribute__((ext_vector_type(8))) float v8f;

// ---------------------------------------------------------------------------
// Stage 1: x [32,64,64,64] -> 4 coarse Haar bands at 4x4, layout [band][pq][i=c][b]
//   ll3[i][j] = (1/8) * sum of 8x8 block  (3 levels of ll-only Haar)
//   band step: ll4/lh/hl/hh = Haar(ll3) * 0.5
// ---------------------------------------------------------------------------
__global__ void k_coarse(const float* __restrict__ x, float* __restrict__ coefA) {
    int bc = blockIdx.x;            // 0..2047
    int b  = bc >> 6;
    int c  = bc & 63;
    const float4* xv = reinterpret_cast<const float4*>(x + (size_t)bc * 4096);

    __shared__ float tile[4096];
    __shared__ float ll3[64];
    float4* tv = reinterpret_cast<float4*>(tile);
    int t = threadIdx.x;
#pragma unroll
    for (int i = 0; i < 4; ++i) tv[t + i * 256] = xv[t + i * 256];
    __syncthreads();

    if (t < 64) {
        int i = t >> 3, j = t & 7;
        float s = 0.f;
#pragma unroll
        for (int r = 0; r < 8; ++r)
#pragma unroll
            for (int cc = 0; cc < 8; ++cc)
                s += tile[(i * 8 + r) * 64 + j * 8 + cc];
        ll3[t] = s * 0.125f;
    }
    __syncthreads();

    if (t < 16) {
        int p = t >> 2, q = t & 3;
        float a  = ll3[(2 * p) * 8 + 2 * q];
        float bb = ll3[(2 * p) * 8 + 2 * q + 1];
        float cv = ll3[(2 * p + 1) * 8 + 2 * q];
        float d  = ll3[(2 * p + 1) * 8 + 2 * q + 1];
        int base = t * 2048 + c * 32 + b;        // [pq][i][b]
        coefA[0 * 32768 + base] = (a + bb + cv + d) * 0.5f;
        coefA[1 * 32768 + base] = (a + bb - cv - d) * 0.5f;
        coefA[2 * 32768 + base] = (a - bb + cv - d) * 0.5f;
        coefA[3 * 32768 + base] = (a - bb - cv + d) * 0.5f;
    }
}

// ---------------------------------------------------------------------------
// Stage 2: per coarse pixel pq, per band t:
//   E_t[b,o] = sum_j G2[b,j] * (coef_t[b,:] @ W_tj[:,o])    (gate folded into A)
// 16 blocks x 256 threads (8 waves; each wave owns one 16x16 D tile).
// Weight tiles (64x64, 16KB) double-buffered through LDS; next expert's tile is
// loaded into registers while the current expert's 16 WMMAs run out of LDS.
// ---------------------------------------------------------------------------
__global__ void k_mix(const float* __restrict__ coefA,
                      const float* __restrict__ WL,
                      const float* __restrict__ WH,
                      const float* __restrict__ lam,
                      float* __restrict__ E) {
    int pq  = blockIdx.x;           // 0..15
    int tid = threadIdx.x;
    int w = tid >> 5, lane = tid & 31;
    int mt = w >> 2, nt = w & 3;
    int l15 = lane & 15, lhalf = lane >> 4;
    int bA = mt * 16 + l15;         // A-matrix row (batch) held by this lane
    int o  = nt * 16 + l15;         // B/D column (output channel)

    __shared__ float Wbuf[2][4096]; // [k][o] weight tile, double buffered

    // gate per A-row, PERM folded: G2[b,4]+=g[b,6], G2[b,5]+=g[b,7]
    float g2[6];
#pragma unroll
    for (int j = 0; j < 6; ++j) g2[j] = lam[bA * 8 + j];
    g2[4] += lam[bA * 8 + 6];
    g2[5] += lam[bA * 8 + 7];

    for (int t = 0; t < 4; ++t) {
        // A fragments for this band into registers (ISA 7.12.2 16x4 f32 layout)
        const float* cb = coefA + (t * 16 + pq) * 2048;
        v2f a[16];
#pragma unroll
        for (int s = 0; s < 16; ++s) {
            int kr = 4 * s + 2 * lhalf;
            a[s].x = cb[kr * 32 + bA];
            a[s].y = cb[(kr + 1) * 32 + bA];
        }

        // prologue: stage expert 0 weight tile
        {
            const float* Wp = (t == 0) ? WL : (WH + (size_t)(t - 1) * 65536);
#pragma unroll
            for (int r = 0; r < 16; ++r)
                Wbuf[0][tid + 256 * r] = Wp[(tid + 256 * r) * 16 + pq];
        }
        __syncthreads();

        v8f acc = {};
        for (int j = 0; j < 6; ++j) {
            // issue next expert's tile loads (kept in flight behind the WMMAs)
            float tmp[16];
            if (j < 5) {
                const float* Wp = (t == 0) ? (WL + (size_t)(j + 1) * 65536)
                                           : (WH + (size_t)((j + 1) * 3 + t - 1) * 65536);
#pragma unroll
                for (int r = 0; r < 16; ++r)
                    tmp[r] = Wp[(tid + 256 * r) * 16 + pq];
            }

            const float* wb = Wbuf[j & 1];
            float g = g2[j];
#pragma unroll
            for (int s = 0; s < 16; ++s) {
                int kr = 4 * s + 2 * lhalf;
                v2f av, bv;
                av.x = a[s].x * g;
                av.y = a[s].y * g;
                bv.x = wb[kr * 64 + o];
                bv.y = wb[(kr + 1) * 64 + o];
                acc = __builtin_amdgcn_wmma_f32_16x16x4_f32(
                    false, av, false, bv, (short)0, acc, false, false);
            }

            if (j < 5) {
#pragma unroll
                for (int r = 0; r < 16; ++r)
                    Wbuf[(j + 1) & 1][tid + 256 * r] = tmp[r];
            }
            __syncthreads();
        }

        // store E_t [t][pq][b][o] (D-tile row layout: VGPR v -> M=v / v+8 by lane half)
#pragma unroll
        for (int v = 0; v < 8; ++v) {
            int brow = mt * 16 + v + lhalf * 8;
            E[((t * 16 + pq) * 32 + brow) * 64 + o] = acc[v];
        }
        __syncthreads();
    }
}

// ---------------------------------------------------------------------------
// Stage 3: Delta_t = E_t - G[b]*coef_t; one inverse Haar step -> R [B][C][8][8].
// ---------------------------------------------------------------------------
__global__ void k_inv(const float* __restrict__ coefA,
                      const float* __restrict__ E,
                      const float* __restrict__ lam,
                      float* __restrict__ R) {
    int idx = blockIdx.x * 256 + threadIdx.x;   // 32768 = 16*32*64
    int o  = idx & 63;
    int b  = (idx >> 6) & 31;
    int pq = idx >> 11;
    int p = pq >> 2, q = pq & 3;
    float G = 0.f;
#pragma unroll
    for (int k = 0; k < 8; ++k) G += lam[b * 8 + k];
    float d[4];
#pragma unroll
    for (int t = 0; t < 4; ++t)
        d[t] = E[((t * 16 + pq) * 32 + b) * 64 + o]
             - G * coefA[(t * 16 + pq) * 2048 + o * 32 + b];
    float r00 = (d[0] + d[1] + d[2] + d[3]) * 0.5f;
    float r01 = (d[0] + d[1] - d[2] - d[3]) * 0.5f;
    float r10 = (d[0] - d[1] + d[2] - d[3]) * 0.5f;
    float r11 = (d[0] - d[1] - d[2] + d[3]) * 0.5f;
    size_t base = ((size_t)(b * 64 + o) * 8 + 2 * p) * 8 + 2 * q;
    R[base]     = r00;
    R[base + 1] = r01;
    R[base + 8] = r10;
    R[base + 9] = r11;
}

// ---------------------------------------------------------------------------
// Stage 4: out = G[b]*x + (1/8)*R broadcast over 8x8 blocks. float4 streaming.
// ---------------------------------------------------------------------------
__global__ void k_out(const float* __restrict__ x,
                      const float* __restrict__ lam,
                      const float* __restrict__ R,
                      float* __restrict__ out) {
    int idx = blockIdx.x * 256 + threadIdx.x;   // float4 index, 2^21 total
    int b  = idx >> 16;
    int c  = (idx >> 10) & 63;
    int h  = (idx >> 4) & 63;
    int w4 = idx & 15;
    float G = 0.f;
#pragma unroll
    for (int k = 0; k < 8; ++k) G += lam[b * 8 + k];
    float4 xv = reinterpret_cast<const float4*>(x)[idx];
    float r = 0.125f * R[((b * 64 + c) * 8 + (h >> 3)) * 8 + (w4 >> 1)];
    float4 ov;
    ov.x = G * xv.x + r;
    ov.y = G * xv.y + r;
    ov.z = G * xv.z + r;
    ov.w = G * xv.w + r;
    reinterpret_cast<float4*>(out)[idx] = ov;
}

extern "C" void kernel_launch(void* const* d_in, const int* in_sizes, int n_in,
                              void* d_out, int out_size, void* d_ws, size_t ws_size,
                              hipStream_t stream) {
    const float* x   = (const float*)d_in[0];   // [32,64,64,64]
    const float* lam = (const float*)d_in[1];   // [32,1,8,1]
    const float* WL  = (const float*)d_in[2];   // [8,64,64,4,4]
    const float* WH  = (const float*)d_in[3];   // [8,3,64,64,4,4]
    float* out = (float*)d_out;

    float* coefA = (float*)d_ws;                // 131072 floats
    float* E     = coefA + 131072;              // 131072 floats
    float* R     = E + 131072;                  // 131072 floats

    k_coarse<<<2048, 256, 0, stream>>>(x, coefA);
    k_mix<<<16, 256, 0, stream>>>(coefA, WL, WH, lam, E);
    k_inv<<<128, 256, 0, stream>>>(coefA, E, lam, R);
    k_out<<<8192, 256, 0, stream>>>(x, lam, R, out);
}